// Radon_13469017440938
// MI455X (gfx1250) — compile-verified
//
#include <hip/hip_runtime.h>

typedef __attribute__((ext_vector_type(2))) float v2f;
typedef __attribute__((ext_vector_type(8))) float v8f;

#define N_ANGLES 180
#define IMG      256
#define SPAD     363          // ceil(sqrt(2)*256)
#define PADB     53           // pad_before = (363-256)/2
#define CXF      181.0f       // (SPAD-1)/2
#define STEPF    (2.0f / 362.0f)
#define BLOCK    512
#define NWAVES   (BLOCK / 32)

// Fetch from LDS-resident unpadded image; zero outside [0,256)^2.
// Equivalent to reference's zero-pad + validity mask + clipped gather.
__device__ __forceinline__ float lds_fetch(const float* simg, int u, int v) {
    bool ok = ((unsigned)u < (unsigned)IMG) & ((unsigned)v < (unsigned)IMG);
    int uc = u < 0 ? 0 : (u > IMG - 1 ? IMG - 1 : u);
    int vc = v < 0 ? 0 : (v > IMG - 1 ? IMG - 1 : v);
    float val = simg[vc * IMG + uc];
    return ok ? val : 0.0f;
}

__device__ __forceinline__ float bilin(const float* simg, float ix, float iy) {
    float x0f = floorf(ix);
    float y0f = floorf(iy);
    float wx = ix - x0f;
    float wy = iy - y0f;
    int u0 = (int)x0f - PADB;   // padded coords -> unpadded image coords
    int v0 = (int)y0f - PADB;
    float v00 = lds_fetch(simg, u0,     v0);
    float v01 = lds_fetch(simg, u0 + 1, v0);      // +x neighbor (wx weight)
    float v10 = lds_fetch(simg, u0,     v0 + 1);  // +y neighbor (wy weight)
    float v11 = lds_fetch(simg, u0 + 1, v0 + 1);
    float top = v00 + wx * (v01 - v00);
    float bot = v10 + wx * (v11 - v10);
    return top + wy * (bot - top);
}

__global__ __launch_bounds__(BLOCK)
void radon_wmma_kernel(const float* __restrict__ img, float* __restrict__ out) {
    extern __shared__ float simg[];   // 256*256 floats = 256 KB of the WGP's 320 KB LDS

    const int b  = blockIdx.x / N_ANGLES;
    const int th = blockIdx.x % N_ANGLES;
    const int tid = threadIdx.x;

    // ---- Stage image into LDS via gfx1250 async-to-LDS path (ASYNCcnt) ----
    {
        const unsigned long long gbase =
            (unsigned long long)(const void*)(img + (size_t)b * IMG * IMG);
        const unsigned lds_base = (unsigned)(unsigned long long)(const void*)simg;
        #pragma unroll 4
        for (int i = tid; i < (IMG * IMG) / 4; i += BLOCK) {
            const unsigned voff = (unsigned)i * 16u;       // byte offset, 16B granules
            const unsigned lds_addr = lds_base + voff;
            asm volatile("global_load_async_to_lds_b128 %0, %1, %2"
                         :: "v"(lds_addr), "v"(voff), "s"(gbase)
                         : "memory");
        }
#if __has_builtin(__builtin_amdgcn_s_wait_asynccnt)
        __builtin_amdgcn_s_wait_asynccnt(0);
#else
        asm volatile("s_wait_asynccnt 0x0" ::: "memory");
#endif
    }
    __syncthreads();

    const float theta = (float)th * 0.017453292519943295f;  // pi/180
    const float ct = cosf(theta);
    const float st = sinf(theta);

    const int lane = tid & 31;
    const int wave = tid >> 5;
    const int m    = lane & 15;         // A-matrix row M (x position within tile)
    const int kb   = (lane >> 4) << 1;  // K offset: lanes 0-15 -> K=0,1 ; 16-31 -> K=2,3

    const v2f ones = {1.0f, 1.0f};      // all-ones B (4x16): D[m][n] = C + sum_k A[m][k]
    float* const obase = out + (size_t)b * SPAD * N_ANGLES + th;  // uniform base

    // 23 x-tiles of 16 cover SPAD=363; tile loop condition is wave-uniform.
    for (int tile = wave; tile * 16 < SPAD; tile += NWAVES) {
        const int   x   = tile * 16 + m;                   // may exceed SPAD-1; rows not stored
        const float lx  = fmaf((float)x, STEPF, -1.0f);
        const float cLx = ct * lx;                         // gx = ct*lx + st*ly
        const float sLx = -st * lx;                        // gy = -st*lx + ct*ly

        auto sample = [&](int y) -> float {
            const float ly = fmaf((float)y, STEPF, -1.0f);
            const float ix = (fmaf(st, ly, cLx) + 1.0f) * CXF;
            const float iy = (fmaf(ct, ly, sLx) + 1.0f) * CXF;
            return bilin(simg, ix, iy);
        };

        v8f acc = {};
        // Main loop: y0 = 0..356 -> every k in [y0, y0+3] is in range; no bounds selects.
        #pragma unroll 2
        for (int y0 = 0; y0 < SPAD - 3; y0 += 4) {
            v2f a = {sample(y0 + kb), sample(y0 + kb + 1)};
            acc = __builtin_amdgcn_wmma_f32_16x16x4_f32(
                false, a, false, ones, (short)0, acc, false, false);
        }
        // Peeled tail: y0 = 360 -> y in {360,361,362,363}; only 363 is invalid.
        {
            const int y = (SPAD - 3) + kb;                 // 360 or 362
            float s0 = sample(y);
            float s1 = sample(y + 1);                      // safe even when y+1 == 363
            s1 = (y + 1 < SPAD) ? s1 : 0.0f;
            v2f a = {s0, s1};
            acc = __builtin_amdgcn_wmma_f32_16x16x4_f32(
                false, a, false, ones, (short)0, acc, false, false);
        }

        // All 16 columns of D are identical. Lane j (j<8) holds D[j][j]; lane 16+j holds
        // D[8+j][j]. Select acc[j] with a cndmask tree and do ONE exec-masked store.
        float val = acc[0];
        #pragma unroll
        for (int r = 1; r < 8; ++r) val = (m == r) ? acc[r] : val;
        const int row = ((lane >> 4) << 3) + m;            // 0..7 (low half) / 8..15 (high half)
        const int xx  = tile * 16 + row;
        if ((m < 8) && (xx < SPAD))
            obase[xx * N_ANGLES] = val;
    }
}

extern "C" void kernel_launch(void* const* d_in, const int* in_sizes, int n_in,
                              void* d_out, int out_size, void* d_ws, size_t ws_size,
                              hipStream_t stream) {
    const float* img = (const float*)d_in[0];
    float* out = (float*)d_out;
    const int B = in_sizes[0] / (IMG * IMG);   // = 2
    dim3 grid(B * N_ANGLES);
    dim3 block(BLOCK);
    size_t shmem = (size_t)IMG * IMG * sizeof(float);  // 256 KB dynamic LDS
    radon_wmma_kernel<<<grid, block, shmem, stream>>>(img, out);
}